// _TorchHashEncoding_69406671503660
// MI455X (gfx1250) — compile-verified
//
#include <hip/hip_runtime.h>
#include <cstdint>
#include <cmath>

// ---------------------------------------------------------------------------
// Multiresolution hash-grid encoding (instant-NGP style), CDNA5 / gfx1250.
// N_LEVELS=16, F=2, LOG2_T=19, BASE_RES=16, SCALE=1.5, N_POINTS=2^21.
// Gather-bound kernel: levels 0..3 staged in LDS via async global->LDS copies
// (ASYNCcnt path), levels 4..15 gathered from L2 (table ~31MB << 192MB L2).
// HBM floor: 256MB out + 16MB in ~= 11.7us @ 23.3TB/s.
// ---------------------------------------------------------------------------

#define NLV 16
#define NLDS 4                      // levels resident in LDS (compile-time)
#define TMASK ((1u << 19) - 1u)
#define LDS_FLOATS 11264            // 44 KB table cache (levels 0..3 ~= 40 KB)
#define BLOCK 256

typedef __attribute__((ext_vector_type(4))) int v4i;

struct LevelTab {
    int res[NLV];
    int off[NLV];     // entry offset of level start
    int hsize[NLV];   // entries in this level (aligned / capped)
};

// ---- async global -> LDS copy (16 B), gfx1250 ASYNCcnt path ---------------
__device__ __forceinline__ void async_g2l_b128(const float* g, float* l) {
#if __has_builtin(__builtin_amdgcn_global_load_async_to_lds_b128)
    __builtin_amdgcn_global_load_async_to_lds_b128(
        (v4i __attribute__((address_space(1))) *)g,
        (v4i __attribute__((address_space(3))) *)l,
        0, 0);
#else
    unsigned lds_off = (unsigned)(uintptr_t)l;            // addr[31:0] = LDS byte addr
    unsigned long long ga = (unsigned long long)(uintptr_t)g;
    asm volatile("global_load_async_to_lds_b128 %0, %1, off"
                 :: "v"(lds_off), "v"(ga) : "memory");
#endif
}

__device__ __forceinline__ void wait_asynccnt0() {
#if __has_builtin(__builtin_amdgcn_s_wait_asynccnt)
    __builtin_amdgcn_s_wait_asynccnt(0);
#else
    asm volatile("s_wait_asynccnt 0" ::: "memory");
#endif
}

__global__ void __launch_bounds__(BLOCK)
hashgrid_encode_kernel(const float* __restrict__ xy,
                       const float* __restrict__ params,
                       float* __restrict__ out,
                       LevelTab lt, int cachedFloats, int npoints)
{
    __shared__ float s_table[LDS_FLOATS];

    const int tid = threadIdx.x;

    // ---- stage small levels into LDS with async copies (multiple of 16 B) --
    const int nv4 = cachedFloats >> 2;
    for (int i = tid; i < nv4; i += BLOCK)
        async_g2l_b128(params + 4 * i, &s_table[4 * i]);
    wait_asynccnt0();
    // Compiler barrier: the async-to-LDS builtin's write to s_table must not
    // be "proven absent" (which would fold the ds_load gathers to poison).
    asm volatile("" ::: "memory");
    __syncthreads();

    const int p = blockIdx.x * BLOCK + tid;
    if (p >= npoints) return;

    const float2 q = reinterpret_cast<const float2*>(xy)[p];
    const float2* __restrict__ tab = reinterpret_cast<const float2*>(params);

    float o[2 * NLV];

#pragma unroll
    for (int l = 0; l < NLV; ++l) {
        const int res   = lt.res[l];
        const int off   = lt.off[l];
        const int hsize = lt.hsize[l];
        const float rf  = (float)(res - 1);

        // pos = xy*(res-1) + 0.5 with NO fp-contraction (must match reference
        // rounding exactly so floorf() picks the same cell)
        const float px = __fadd_rn(__fmul_rn(q.x, rf), 0.5f);
        const float py = __fadd_rn(__fmul_rn(q.y, rf), 0.5f);
        const float fx = floorf(px), fy = floorf(py);
        const float wx = px - fx,    wy = py - fy;
        const int gx = (int)fx, gy = (int)fy;

        int i00, i10, i01, i11;
        if (hsize >= res * res) {                 // dense level
            const int b = gx + gy * res;
            i00 = b;        i10 = b + 1;
            i01 = b + res;  i11 = b + res + 1;
        } else {                                  // hashed level (hsize == 2^19)
            const unsigned hx0 = (unsigned)gx;
            const unsigned hx1 = hx0 + 1u;
            const unsigned hy0 = (unsigned)gy * 2654435761u;
            const unsigned hy1 = ((unsigned)gy + 1u) * 2654435761u;
            i00 = (int)((hx0 ^ hy0) & TMASK);
            i10 = (int)((hx1 ^ hy0) & TMASK);
            i01 = (int)((hx0 ^ hy1) & TMASK);
            i11 = (int)((hx1 ^ hy1) & TMASK);
        }
        i00 += off; i10 += off; i01 += off; i11 += off;

        float2 f00, f10, f01, f11;
        if (l < NLDS) {                           // LDS-resident level (static)
            f00 = *reinterpret_cast<const float2*>(&s_table[2 * i00]);
            f10 = *reinterpret_cast<const float2*>(&s_table[2 * i10]);
            f01 = *reinterpret_cast<const float2*>(&s_table[2 * i01]);
            f11 = *reinterpret_cast<const float2*>(&s_table[2 * i11]);
        } else {                                  // L2-resident gather
            f00 = tab[i00]; f10 = tab[i10]; f01 = tab[i01]; f11 = tab[i11];
        }

        const float w00 = (1.0f - wx) * (1.0f - wy);
        const float w10 = wx * (1.0f - wy);
        const float w01 = (1.0f - wx) * wy;
        const float w11 = wx * wy;

        o[2 * l + 0] = fmaf(f11.x, w11, fmaf(f01.x, w01, fmaf(f10.x, w10, f00.x * w00)));
        o[2 * l + 1] = fmaf(f11.y, w11, fmaf(f01.y, w01, fmaf(f10.y, w10, f00.y * w00)));
    }

    // 32 contiguous floats per point -> 8x b128 stores; each wave writes a
    // fully-dirty contiguous 4KB region (HBM-efficient).
    float4* __restrict__ dst = reinterpret_cast<float4*>(out + (size_t)p * (2 * NLV));
    const float4* src = reinterpret_cast<const float4*>(o);
#pragma unroll
    for (int i = 0; i < 8; ++i) dst[i] = src[i];
}

// ---------------------------------------------------------------------------
// Host side: replicate the reference's level-geometry math bit-for-bit
// (pow(2.0, l*log2(1.5)) in double, like CPython's 2.0**x).
// ---------------------------------------------------------------------------
static void build_level_tab(LevelTab& lt) {
    const double log2s = std::log2(1.5);
    int off = 0;
    for (int l = 0; l < NLV; ++l) {
        const double scale = std::pow(2.0, (double)l * log2s) * 16.0 - 1.0;
        const int r = (int)(std::ceil(scale) + 1.0);
        long long sz = ((long long)r * r + 7) / 8 * 8;
        if (sz > (1LL << 19)) sz = 1LL << 19;
        lt.res[l] = r;
        lt.off[l] = off;
        lt.hsize[l] = (int)sz;
        off += (int)sz;
    }
}

extern "C" void kernel_launch(void* const* d_in, const int* in_sizes, int n_in,
                              void* d_out, int out_size, void* d_ws, size_t ws_size,
                              hipStream_t stream) {
    const float* xy     = (const float*)d_in[0];
    const float* params = (const float*)d_in[1];
    float* out          = (float*)d_out;

    LevelTab lt;
    build_level_tab(lt);

    // Floats to stage into LDS = everything below level NLDS (always fits:
    // worst-case levels 0..3 total < 5.3K entries = 42KB <= 44KB buffer).
    int cachedFloats = 2 * (lt.off[NLDS - 1] + lt.hsize[NLDS - 1]);
    if (cachedFloats > LDS_FLOATS) cachedFloats = LDS_FLOATS;  // safety clamp
    cachedFloats &= ~3;   // b128 copies only (level sizes are 8-entry aligned)

    const int npoints = in_sizes[0] / 2;
    const int grid = (npoints + BLOCK - 1) / BLOCK;

    hashgrid_encode_kernel<<<grid, BLOCK, 0, stream>>>(
        xy, params, out, lt, cachedFloats, npoints);
}